// GNN_12189117186811
// MI455X (gfx1250) — compile-verified
//
#include <hip/hip_runtime.h>

typedef float v2f __attribute__((ext_vector_type(2)));
typedef float v8f __attribute__((ext_vector_type(8)));

// ---------------------------------------------------------------------------
// Degree / normalization
// ---------------------------------------------------------------------------
__global__ void __launch_bounds__(256) gcn_init_deg(float* __restrict__ deg, long long n) {
  long long i = (long long)blockIdx.x * 256 + threadIdx.x;
  if (i < n) deg[i] = 1.0f;  // self-loop contributes 1 to every node's in-degree
}

__global__ void __launch_bounds__(256) gcn_count_deg(const long long* __restrict__ dst,
                                                     float* __restrict__ deg, long long E) {
  long long e = (long long)blockIdx.x * 256 + threadIdx.x;
  if (e < E) atomicAdd(&deg[dst[e]], 1.0f);
}

__global__ void __launch_bounds__(256) gcn_rsqrt(float* __restrict__ deg, long long n) {
  long long i = (long long)blockIdx.x * 256 + threadIdx.x;
  if (i < n) deg[i] = rsqrtf(deg[i]);  // deg >= 1 always (self loops)
}

// ---------------------------------------------------------------------------
// Dense GEMM via V_WMMA_F32_16X16X4_F32: Out[M x NT] = A[M x K] * W[K x NT]
// One wave owns a 16 x NT output strip (NT/16 accumulators): each A fragment is
// loaded once and reused across all N-tiles.
// W is staged through LDS in 32-row K-panels with a K-PAIR-INTERLEAVED layout:
//   element (r, c) -> smW[(r>>1)*QPITCH + 2*c + (r&1)],  QPITCH = 2*NT + 32
// so a B fragment (rows 2q, 2q+1 at column c) is one aligned ds_load_b64 into
// an even VGPR pair (no re-pairing movs). QPITCH % 64 == 32 puts the two lane
// halves in opposite 32-bank groups -> conflict-free b64 reads.
// A-frag (16x4 f32, 2 VGPR): lane L in [0,15] -> row M=L, K={0,1}; lanes 16..31 -> K={2,3}.
// B-frag (4x16 f32, 2 VGPR): VGPR g, lane-half h -> row K=2h+g, col N=lane&15.
// C/D (16x16 f32, 8 VGPR): VGPR v -> row M=v (lanes 0-15) / v+8 (lanes 16-31), col N=lane&15.
// ---------------------------------------------------------------------------
template <int K, int NT>
__global__ void __launch_bounds__(256) gcn_gemm_wmma(const float* __restrict__ A,
                                                     const float* __restrict__ W,
                                                     float* __restrict__ Out,
                                                     int Mtiles) {
  constexpr int KP     = 32;            // K-panel rows staged in LDS
  constexpr int QPITCH = 2 * NT + 32;   // dwords per interleaved K-pair row
  constexpr int NTILES = NT / 16;
  __shared__ float smW[(KP / 2) * QPITCH];

  const int tid     = threadIdx.x;
  const int lane    = tid & 31;
  const int wib     = tid >> 5;                // wave in block (8 waves)
  const int mTile   = blockIdx.x * 8 + wib;
  const bool active = (mTile < Mtiles);        // wave-uniform

  const int m  = lane & 15;
  const int hh = lane >> 4;                    // lane half: 0 or 1

  const float* __restrict__ arow = A + (size_t)(mTile * 16 + m) * K + (size_t)(hh * 2);

  v8f acc[NTILES];
#pragma unroll
  for (int t = 0; t < NTILES; ++t) acc[t] = v8f{0.f, 0.f, 0.f, 0.f, 0.f, 0.f, 0.f, 0.f};

#pragma unroll 1
  for (int p = 0; p < K; p += KP) {
    // cooperative stage: rows [p, p+KP) of W into interleaved LDS (all 256 threads)
#pragma unroll
    for (int idx = tid; idx < KP * NT; idx += 256) {
      int r = idx / NT;           // NT is a power of two -> shift
      int c = idx & (NT - 1);
      smW[(r >> 1) * QPITCH + 2 * c + (r & 1)] = W[(size_t)(p + r) * NT + c];
    }
    __syncthreads();

    if (active) {
#pragma unroll
      for (int kk = 0; kk < KP; kk += 4) {
        v2f a;
        a.x = arow[p + kk];
        a.y = arow[p + kk + 1];
        // rows (kk+2*hh, kk+2*hh+1) at column m, tiles at +32-dword steps
        const float* __restrict__ brow = smW + (kk / 2 + hh) * QPITCH + 2 * m;
#pragma unroll
        for (int t = 0; t < NTILES; ++t) {
          v2f b = *(const v2f*)(brow + 32 * t);   // one ds_load_b64
          // (neg_a, A, neg_b, B, c_mod, C, reuse_a, reuse_b)
          acc[t] = __builtin_amdgcn_wmma_f32_16x16x4_f32(false, a, false, b, (short)0,
                                                         acc[t], false, false);
        }
      }
    }
    __syncthreads();  // protect LDS before next panel overwrite
  }

  if (active) {
    float* __restrict__ obase = Out + (size_t)(mTile * 16 + hh * 8) * NT + m;
#pragma unroll
    for (int t = 0; t < NTILES; ++t) {
#pragma unroll
      for (int v = 0; v < 8; ++v) obase[(size_t)v * NT + t * 16] = acc[t][v];
    }
  }
}

// ---------------------------------------------------------------------------
// Edge aggregation: Out[dst] += dinv[src]*dinv[dst] * H[src], F = 1<<FSHIFT floats/row.
// One thread per (edge, float4 chunk) -> coalesced gather, fp32 atomic scatter.
// Feature matrix (<= 51.2 MB) is L2-resident on MI455X (192 MB L2), so this is
// L2-atomic-throughput bound, not HBM bound.
// ---------------------------------------------------------------------------
template <int FSHIFT>
__global__ void __launch_bounds__(256) gcn_aggregate(const float* __restrict__ H,
                                                     float* __restrict__ Out,
                                                     const long long* __restrict__ src,
                                                     const long long* __restrict__ dst,
                                                     const float* __restrict__ dinv,
                                                     long long total) {
  long long tid = (long long)blockIdx.x * 256 + threadIdx.x;
  if (tid >= total) return;
  const int QSHIFT = FSHIFT - 2;  // float4 chunks per row
  int chunk   = (int)(tid & ((1 << QSHIFT) - 1));
  long long e = tid >> QSHIFT;
  long long s = src[e];
  long long d = dst[e];
  float nrm = dinv[s] * dinv[d];
  const float4 hv = *((const float4*)(H + (s << FSHIFT)) + chunk);
  float* o = Out + (d << FSHIFT) + ((long long)chunk << 2);
  atomicAdd(o + 0, nrm * hv.x);
  atomicAdd(o + 1, nrm * hv.y);
  atomicAdd(o + 2, nrm * hv.z);
  atomicAdd(o + 3, nrm * hv.w);
}

// ---------------------------------------------------------------------------
// Epilogue: Acc[i,f] += dinv[i]^2 * Hself[i,f] (self-loop edge) + bias[f]; optional ReLU.
// ---------------------------------------------------------------------------
template <int FSHIFT, bool RELU>
__global__ void __launch_bounds__(256) gcn_epilogue(float* __restrict__ Acc,
                                                    const float* __restrict__ Hself,
                                                    const float* __restrict__ dinv,
                                                    const float* __restrict__ bias,
                                                    long long total) {
  long long tid = (long long)blockIdx.x * 256 + threadIdx.x;
  if (tid >= total) return;
  long long i = tid >> FSHIFT;
  int f = (int)(tid & ((1 << FSHIFT) - 1));
  float di = dinv[i];
  float v = Acc[tid] + di * di * Hself[tid] + bias[f];
  Acc[tid] = RELU ? fmaxf(v, 0.0f) : v;
}

// ---------------------------------------------------------------------------
// Launcher
// ---------------------------------------------------------------------------
extern "C" void kernel_launch(void* const* d_in, const int* in_sizes, int n_in,
                              void* d_out, int out_size, void* d_ws, size_t ws_size,
                              hipStream_t stream) {
  const float*     x  = (const float*)d_in[0];      // [N,128]
  const long long* ei = (const long long*)d_in[1];  // [2,E] int64
  const float*     W1 = (const float*)d_in[2];      // [128,128]
  const float*     b1 = (const float*)d_in[3];      // [128]
  const float*     W2 = (const float*)d_in[4];      // [128,64]
  const float*     b2 = (const float*)d_in[5];      // [64]

  const long long N = (long long)in_sizes[0] / 128;  // 100000 (multiple of 16)
  const long long E = (long long)in_sizes[1] / 2;    // 1600000
  const long long* src = ei;
  const long long* dst = ei + E;

  // Workspace layout (fp32): [deg/dinv: N][h1: N*128 (reused as t: N*64)][agg1: N*128]
  float* deg  = (float*)d_ws;
  float* h1   = deg + N;
  float* agg1 = h1 + (size_t)N * 128;
  float* out  = (float*)d_out;

  auto blk = [](long long threads) { return (unsigned)((threads + 255) / 256); };

  // --- symmetric normalization: dinv = rsqrt(1 + in-degree) ---
  gcn_init_deg<<<blk(N), 256, 0, stream>>>(deg, N);
  gcn_count_deg<<<blk(E), 256, 0, stream>>>(dst, deg, E);
  gcn_rsqrt<<<blk(N), 256, 0, stream>>>(deg, N);

  hipMemsetAsync(agg1, 0, (size_t)N * 128 * sizeof(float), stream);
  hipMemsetAsync(out, 0, (size_t)out_size * sizeof(float), stream);

  const int Mtiles = (int)(N / 16);                      // 6250
  const unsigned gemmBlocks = (unsigned)((Mtiles + 7) / 8);

  // --- layer 1: h1 = x @ W1 ; agg1 = A_hat @ h1 ; relu(+b1) ---
  gcn_gemm_wmma<128, 128><<<gemmBlocks, 256, 0, stream>>>(x, W1, h1, Mtiles);
  gcn_aggregate<7><<<blk(E << 5), 256, 0, stream>>>(h1, agg1, src, dst, deg, E << 5);
  gcn_epilogue<7, true><<<blk(N << 7), 256, 0, stream>>>(agg1, h1, deg, b1, N << 7);

  // --- layer 2: t = agg1 @ W2 (t overwrites h1, which is dead now) ---
  float* t = h1;
  gcn_gemm_wmma<128, 64><<<gemmBlocks, 256, 0, stream>>>(agg1, W2, t, Mtiles);
  gcn_aggregate<6><<<blk(E << 4), 256, 0, stream>>>(t, out, src, dst, deg, E << 4);
  gcn_epilogue<6, false><<<blk(N << 6), 256, 0, stream>>>(out, t, deg, b2, N << 6);
}